// SelfAttention_14164802142491
// MI455X (gfx1250) — compile-verified
//
#include <hip/hip_runtime.h>

// Problem constants from the reference: B=32, L=2048, H=512, f32 throughout.
#define B_ 32
#define L_ 2048
#define H_ 512

typedef __attribute__((ext_vector_type(2))) float v2f;
typedef __attribute__((ext_vector_type(4))) float v4f;
typedef __attribute__((ext_vector_type(8))) float v8f;

// ---------------------------------------------------------------------------
// Pass 1: scores_relu[b][l] = relu( dot(inputs[b,l,:], w) )
// Uses V_WMMA_F32_16X16X4_F32: each wave32 computes 16 L-rows at a time.
//   A (16x4 f32):  lanes 0-15 hold (M=lane, K=0 in v0, K=1 in v1)
//                  lanes 16-31 hold (M=lane-16, K=2 in v0, K=3 in v1)
//   B (4x16 f32):  row K striped across lanes; we place w only in column N=0
//                  (lane 0 -> K=0,1 ; lane 16 -> K=2,3 ; all other lanes 0)
//   D column N=0:  lane 0 v[0..7] = rows M=0..7, lane 16 v[0..7] = rows M=8..15
// Block = 256 threads = 8 waves = 128 rows. Grid = (L/128, B).
// ---------------------------------------------------------------------------
__global__ __launch_bounds__(256) void sa_scores_wmma(
    const float* __restrict__ inputs, const float* __restrict__ w,
    float* __restrict__ scores) {
  const int b    = blockIdx.y;
  const int lane = threadIdx.x & 31;
  const int wave = threadIdx.x >> 5;
  const int l0   = blockIdx.x * 128 + wave * 16;
  const int m    = lane & 15;           // row within the 16-row tile
  const int kh   = (lane >> 4) << 1;    // K offset: 0 (lanes<16) or 2 (lanes>=16)

  const float* rowPtr = inputs + ((size_t)b * L_ + (size_t)(l0 + m)) * H_ + kh;
  const bool  isCol0  = (m == 0);       // lane 0 and lane 16 carry B column N=0

  v8f acc = {};
  for (int h = 0; h < H_; h += 4) {
    v2f a  = *(const v2f*)(rowPtr + h);           // A: 2 consecutive K values
    v2f wv = *(const v2f*)(w + h + kh);           // matching weight chunk
    v2f bv;
    bv.x = isCol0 ? wv.x : 0.0f;                  // B[k][0] = w[h+k], else 0
    bv.y = isCol0 ? wv.y : 0.0f;
    // D = A*B + C, full f32 precision
    acc = __builtin_amdgcn_wmma_f32_16x16x4_f32(
        /*neg_a=*/false, a, /*neg_b=*/false, bv,
        /*c_mod=*/(short)0, acc, /*reuse_a=*/false, /*reuse_b=*/false);
  }

  // Column N=0 of D holds the 16 scores. Apply relu here (softmax of relu).
  if (m == 0) {
    float* s = scores + (size_t)b * L_ + l0 + (lane >> 4) * 8;
#pragma unroll
    for (int i = 0; i < 8; ++i) s[i] = fmaxf(acc[i], 0.0f);
  }
}

// ---------------------------------------------------------------------------
// Pass 2: masked softmax per batch. softmax-then-mask-then-renormalize is
// algebraically a masked softmax (the unmasked denominator cancels), so:
//   attn[l] = (l < len) ? exp(s[l]-max) / sum_{l'<len} exp(s[l']-max) : 0
// relu scores are >= 0 so max over all L is >= 0 automatically.
// One block (256 threads) per batch; LDS tree reductions.
// ---------------------------------------------------------------------------
__global__ __launch_bounds__(256) void sa_attn(
    const float* __restrict__ scores, const int* __restrict__ lengths,
    float* __restrict__ attn_ws, float* __restrict__ attn_out) {
  const int b   = blockIdx.x;
  const int tid = threadIdx.x;
  const int len = lengths[b];
  const float* s = scores + (size_t)b * L_;
  __shared__ float red[256];

  // max over all L of relu(score) (matches reference softmax normalization)
  float mx = 0.0f;
  for (int l = tid; l < L_; l += 256) mx = fmaxf(mx, s[l]);
  red[tid] = mx; __syncthreads();
  for (int o = 128; o > 0; o >>= 1) {
    if (tid < o) red[tid] = fmaxf(red[tid], red[tid + o]);
    __syncthreads();
  }
  mx = red[0]; __syncthreads();

  // masked exp-sum
  float sum = 0.0f;
  for (int l = tid; l < L_; l += 256)
    sum += (l < len) ? __expf(s[l] - mx) : 0.0f;
  red[tid] = sum; __syncthreads();
  for (int o = 128; o > 0; o >>= 1) {
    if (tid < o) red[tid] += red[tid + o];
    __syncthreads();
  }
  const float inv = 1.0f / red[0];   // len >= 1 -> sum > 0

  for (int l = tid; l < L_; l += 256) {
    const float e = (l < len) ? __expf(s[l] - mx) * inv : 0.0f;
    attn_ws[(size_t)b * L_ + l]  = e;   // for pass 3
    attn_out[(size_t)b * L_ + l] = e;   // second tuple output
  }
}

// ---------------------------------------------------------------------------
// Pass 3: weighted[b,l,h] = inputs[b,l,h] * attn[b,l]. Pure streaming,
// float4 per thread (H=512 is a multiple of 4, no row crossing).
// ---------------------------------------------------------------------------
__global__ __launch_bounds__(256) void sa_weighted(
    const float* __restrict__ inputs, const float* __restrict__ attn,
    float* __restrict__ out) {
  const size_t i4  = (size_t)blockIdx.x * 256 + threadIdx.x;
  const size_t idx = i4 * 4;
  const size_t bl  = idx / H_;          // flat b*L + l
  const float  a   = attn[bl];
  v4f v = *(const v4f*)(inputs + idx);
  v4f r = { v.x * a, v.y * a, v.z * a, v.w * a };
  *(v4f*)(out + idx) = r;
}

// ---------------------------------------------------------------------------
// Inputs (setup_inputs order): [0] inputs f32 [B,L,H], [1] lengths i32 [B],
// [2] att_weights f32 [1,H]. Output: weighted [B,L,H] then attn [B,L], flat.
// ---------------------------------------------------------------------------
extern "C" void kernel_launch(void* const* d_in, const int* in_sizes, int n_in,
                              void* d_out, int out_size, void* d_ws, size_t ws_size,
                              hipStream_t stream) {
  const float* inputs  = (const float*)d_in[0];
  const int*   lengths = (const int*)d_in[1];
  const float* w       = (const float*)d_in[2];

  float* weighted = (float*)d_out;
  float* attn_out = (float*)d_out + (size_t)B_ * L_ * H_;

  float* scores  = (float*)d_ws;                  // B*L floats (256 KB)
  float* attn_ws = scores + (size_t)B_ * L_;      // B*L floats (256 KB)

  sa_scores_wmma<<<dim3(L_ / 128, B_), 256, 0, stream>>>(inputs, w, scores);
  sa_attn<<<B_, 256, 0, stream>>>(scores, lengths, attn_ws, attn_out);

  const int n4 = (B_ * L_ * H_) / 4;              // 8,388,608 float4s
  sa_weighted<<<n4 / 256, 256, 0, stream>>>(inputs, attn_ws, weighted);
}